// FullyConnectedVQCs_15t5t1_12077448036875
// MI455X (gfx1250) — compile-verified
//
#include <hip/hip_runtime.h>

// ---------------------------------------------------------------------------
// FullyConnectedVQCs: every circuit is linear in the (real) embedded state, so
// precompute M = Re(U^dag Z U) per circuit (setup kernel), then per batch
// element evaluate quadratic forms s^T M s.  The 32x32 (5-qubit) quadform is
// done with v_wmma_f32_16x16x4_f32 as diag(S*M5*S^T) per 16-row tile.
// ---------------------------------------------------------------------------

#define N_BLOCKS 5
#define M3_PACK 36                     // 8 diag + 28 (2*offdiag)
#define M3_FLOATS (4 * 5 * 3 * M3_PACK) // 2160
#define M5_OFF M3_FLOATS               // f32 [32][32] row-major after M3

typedef float v2f __attribute__((ext_vector_type(2)));
typedef float v8f __attribute__((ext_vector_type(8)));

// ============================ setup kernel =================================
// 1 block x 128 threads.
//  lanes 0..19  : compose one 3q circuit each (stage s=t/5 -> theta_q3[{1,2,3,0}[s]], block b=t%5)
//  lanes 32..63 : evolve basis column (t-32) through the 5q circuit -> LDS
//  all 128      : reduce M5[i][j] = sum_k sgn(k)*Re(conj(U[k][i])*U[k][j])
__global__ __launch_bounds__(128) void vqc_setup(const float* __restrict__ th3,
                                                 const float* __restrict__ th5,
                                                 float* __restrict__ ws) {
  __shared__ float Ur[32][33];
  __shared__ float Ui[32][33];
  const int t = threadIdx.x;

  if (t < 20) {
    const int smap[4] = {1, 2, 3, 0};
    const int s = t / 5, b = t % 5;
    const float* th = th3 + (smap[s] * 5 + b) * 45;  // [5][3][3]
    float ur[8][8], ui[8][8];
#pragma unroll
    for (int i = 0; i < 8; i++)
#pragma unroll
      for (int j = 0; j < 8; j++) { ur[i][j] = (i == j) ? 1.f : 0.f; ui[i][j] = 0.f; }

#pragma unroll
    for (int l = 0; l < 5; l++) {
#pragma unroll
      for (int w = 0; w < 3; w++) {
        const float phi = th[(l * 3 + w) * 3 + 0];
        const float the = th[(l * 3 + w) * 3 + 1];
        const float omg = th[(l * 3 + w) * 3 + 2];
        float ch, sh, cp, sp, cm, sm;
        sincosf(0.5f * the, &sh, &ch);
        sincosf(0.5f * (phi + omg), &sp, &cp);
        sincosf(0.5f * (phi - omg), &sm, &cm);
        // Rot = [[epo*c, -conj(emo)*s],[emo*s, conj(epo)*c]], epo=cp-i*sp, emo=cm-i*sm
        const float g00r = cp * ch,  g00i = -sp * ch;
        const float g01r = -cm * sh, g01i = -sm * sh;
        const float g10r = cm * sh,  g10i = -sm * sh;
        const float g11r = cp * ch,  g11i = sp * ch;
        const int str = 4 >> w;  // wire w acts on bit (2-w)
#pragma unroll
        for (int i0 = 0; i0 < 8; i0++) {
          if ((i0 & str) == 0) {
            const int i1 = i0 | str;
#pragma unroll
            for (int c = 0; c < 8; c++) {
              const float xr = ur[i0][c], xi = ui[i0][c];
              const float yr = ur[i1][c], yi = ui[i1][c];
              ur[i0][c] = g00r * xr - g00i * xi + g01r * yr - g01i * yi;
              ui[i0][c] = g00r * xi + g00i * xr + g01r * yi + g01i * yr;
              ur[i1][c] = g10r * xr - g10i * xi + g11r * yr - g11i * yi;
              ui[i1][c] = g10r * xi + g10i * xr + g11r * yi + g11i * yr;
            }
          }
        }
      }
      const int r = l % 2 + 1;
#pragma unroll
      for (int i = 0; i < 3; i++) {
        const int tt = (i + r) % 3;
        const int cmk = 1 << (2 - i), tmk = 1 << (2 - tt);
#pragma unroll
        for (int k = 0; k < 8; k++) {
          if ((k & cmk) && !(k & tmk)) {
            const int k2 = k | tmk;
#pragma unroll
            for (int c = 0; c < 8; c++) {
              float tr = ur[k][c]; ur[k][c] = ur[k2][c]; ur[k2][c] = tr;
              float ti = ui[k][c]; ui[k][c] = ui[k2][c]; ui[k2][c] = ti;
            }
          }
        }
      }
    }
    // M_w = Re(U^dag Z_w U), packed symmetric (diag, then 2*offdiag i<j)
#pragma unroll
    for (int w = 0; w < 3; w++) {
      float* dst = ws + ((s * 5 + b) * 3 + w) * M3_PACK;
      const int sb = 2 - w;
      float M[8][8];
#pragma unroll
      for (int i = 0; i < 8; i++)
#pragma unroll
        for (int j = 0; j < 8; j++) {
          if (j >= i) {
            float acc = 0.f;
#pragma unroll
            for (int k = 0; k < 8; k++) {
              const float sg = ((k >> sb) & 1) ? -1.f : 1.f;
              acc += sg * (ur[k][i] * ur[k][j] + ui[k][i] * ui[k][j]);
            }
            M[i][j] = acc;
          }
        }
#pragma unroll
      for (int i = 0; i < 8; i++) dst[i] = M[i][i];
      int pos = 8;
#pragma unroll
      for (int i = 0; i < 8; i++)
#pragma unroll
        for (int j = i + 1; j < 8; j++) dst[pos++] = 2.f * M[i][j];
    }
  } else if (t >= 32 && t < 64) {
    const int col = t - 32;
    float vr[32], vi[32];
#pragma unroll
    for (int k = 0; k < 32; k++) { vr[k] = (k == col) ? 1.f : 0.f; vi[k] = 0.f; }
#pragma unroll
    for (int l = 0; l < 5; l++) {
#pragma unroll
      for (int w = 0; w < 5; w++) {
        const float phi = th5[(l * 5 + w) * 3 + 0];
        const float the = th5[(l * 5 + w) * 3 + 1];
        const float omg = th5[(l * 5 + w) * 3 + 2];
        float ch, sh, cp, sp, cm, sm;
        sincosf(0.5f * the, &sh, &ch);
        sincosf(0.5f * (phi + omg), &sp, &cp);
        sincosf(0.5f * (phi - omg), &sm, &cm);
        const float g00r = cp * ch,  g00i = -sp * ch;
        const float g01r = -cm * sh, g01i = -sm * sh;
        const float g10r = cm * sh,  g10i = -sm * sh;
        const float g11r = cp * ch,  g11i = sp * ch;
        const int str = 16 >> w;
#pragma unroll
        for (int i0 = 0; i0 < 32; i0++) {
          if ((i0 & str) == 0) {
            const int i1 = i0 | str;
            const float xr = vr[i0], xi = vi[i0], yr = vr[i1], yi = vi[i1];
            vr[i0] = g00r * xr - g00i * xi + g01r * yr - g01i * yi;
            vi[i0] = g00r * xi + g00i * xr + g01r * yi + g01i * yr;
            vr[i1] = g10r * xr - g10i * xi + g11r * yr - g11i * yi;
            vi[i1] = g10r * xi + g10i * xr + g11r * yi + g11i * yr;
          }
        }
      }
      const int r = l % 4 + 1;
#pragma unroll
      for (int i = 0; i < 5; i++) {
        const int tt = (i + r) % 5;
        const int cmk = 1 << (4 - i), tmk = 1 << (4 - tt);
#pragma unroll
        for (int k = 0; k < 32; k++) {
          if ((k & cmk) && !(k & tmk)) {
            const int k2 = k | tmk;
            float tr = vr[k]; vr[k] = vr[k2]; vr[k2] = tr;
            float ti = vi[k]; vi[k] = vi[k2]; vi[k2] = ti;
          }
        }
      }
    }
#pragma unroll
    for (int k = 0; k < 32; k++) { Ur[k][col] = vr[k]; Ui[k][col] = vi[k]; }
  }
  __syncthreads();
  // M5[i][j] = sum_k sgn(k)*(Ur[k][i]Ur[k][j] + Ui[k][i]Ui[k][j]); Z on wire 0 = bit4
  for (int e = t; e < 1024; e += 128) {
    const int i = e >> 5, j = e & 31;
    float acc = 0.f;
    for (int k = 0; k < 32; k++) {
      const float sg = (k & 16) ? -1.f : 1.f;
      acc += sg * (Ur[k][i] * Ur[k][j] + Ui[k][i] * Ui[k][j]);
    }
    ws[M5_OFF + e] = acc;
  }
}

// ============================ main kernel ==================================
__device__ __forceinline__ void buildP(float a0, float a1, float a2, float* p) {
  float c0, s0, c1, s1, c2, s2;
  sincosf(0.5f * a0, &s0, &c0);
  sincosf(0.5f * a1, &s1, &c1);
  sincosf(0.5f * a2, &s2, &c2);
  float v[8];
  v[0] = c0 * c1 * c2; v[1] = c0 * c1 * s2; v[2] = c0 * s1 * c2; v[3] = c0 * s1 * s2;
  v[4] = s0 * c1 * c2; v[5] = s0 * c1 * s2; v[6] = s0 * s1 * c2; v[7] = s0 * s1 * s2;
#pragma unroll
  for (int i = 0; i < 8; i++) p[i] = v[i] * v[i];
  int pos = 8;
#pragma unroll
  for (int i = 0; i < 8; i++)
#pragma unroll
    for (int j = i + 1; j < 8; j++) p[pos++] = v[i] * v[j];
}

__device__ __forceinline__ float dot36(const float* p, const float* m) {
  float acc = 0.f;
#pragma unroll
  for (int i = 0; i < 36; i += 4) {
    const float4 mv = *(const float4*)(m + i);
    acc = fmaf(p[i], mv.x, fmaf(p[i + 1], mv.y, fmaf(p[i + 2], mv.z, fmaf(p[i + 3], mv.w, acc))));
  }
  return acc;
}

__global__ __launch_bounds__(256) void vqc_main(const float* __restrict__ x,
                                                const float* __restrict__ ws,
                                                float* __restrict__ out) {
  __shared__ __attribute__((aligned(16))) float M3p[M3_FLOATS];
  __shared__ float M5[32][33];
  __shared__ float S[8][32][33];
  const int tid = threadIdx.x;

  for (int i = tid; i < M3_FLOATS; i += 256) M3p[i] = ws[i];
  for (int i = tid; i < 1024; i += 256) M5[i >> 5][i & 31] = ws[M5_OFF + i];
  __syncthreads();

  const int gid = blockIdx.x * 256 + tid;
  float h[15];
  h[0] = 0.f; h[14] = 0.f;
  const float* xp = x + (long long)gid * 13;
#pragma unroll
  for (int i = 0; i < 13; i++) h[1 + i] = xp[i];

  // three 3->3 stages (theta_q3[1],[2],[3]); stages 1,2 use the 5x3-transpose shuffle
#pragma unroll
  for (int s = 0; s < 3; s++) {
    float hn[15];
#pragma unroll
    for (int b = 0; b < 5; b++) {
      float a[3];
#pragma unroll
      for (int k = 0; k < 3; k++) {
        const int idx = 3 * b + k;
        a[k] = (s == 0) ? h[idx] : h[3 * (idx % 5) + idx / 5];
      }
      float p[36];
      buildP(a[0], a[1], a[2], p);
#pragma unroll
      for (int w = 0; w < 3; w++)
        hn[3 * b + w] = dot36(p, &M3p[((s * 5 + b) * 3 + w) * M3_PACK]);
    }
#pragma unroll
    for (int i = 0; i < 15; i++) h[i] = hn[i];
  }
  // final 3->1 stage (theta_q3[0])
  float a5[5];
#pragma unroll
  for (int b = 0; b < 5; b++) {
    float p[36];
    buildP(h[3 * b], h[3 * b + 1], h[3 * b + 2], p);
    a5[b] = dot36(p, &M3p[((3 * 5 + b) * 3 + 0) * M3_PACK]);
  }

  // 5-qubit embedding: s32[k] = prod_w t_w[bit_{4-w}(k)]
  float c[5], sn[5];
#pragma unroll
  for (int w = 0; w < 5; w++) sincosf(0.5f * a5[w], &sn[w], &c[w]);
  const int wave = tid >> 5, lane = tid & 31;
  float (*Sw)[33] = S[wave];
  {
    float e2[4], e3[8], e4[16];
#pragma unroll
    for (int i = 0; i < 4; i++) e2[i] = ((i >> 1) ? sn[0] : c[0]) * ((i & 1) ? sn[1] : c[1]);
#pragma unroll
    for (int i = 0; i < 8; i++) e3[i] = e2[i >> 1] * ((i & 1) ? sn[2] : c[2]);
#pragma unroll
    for (int i = 0; i < 16; i++) e4[i] = e3[i >> 1] * ((i & 1) ? sn[3] : c[3]);
#pragma unroll
    for (int k = 0; k < 16; k++) {
      Sw[lane][2 * k]     = e4[k] * c[4];
      Sw[lane][2 * k + 1] = e4[k] * sn[4];
    }
  }
  __syncthreads();

  // out[r] = diag(S * M5 * S^T)[r] via v_wmma_f32_16x16x4_f32; 2 passes of 16 rows.
  const float MULT = (float)(3.141592653589793 - 1.1920928955078125e-07);
  const int L = lane & 15, hf = lane >> 4;
  const int gbase = blockIdx.x * 256 + wave * 32;
#pragma unroll
  for (int p5 = 0; p5 < 2; p5++) {
    v8f acc0 = {0.f, 0.f, 0.f, 0.f, 0.f, 0.f, 0.f, 0.f};
    v8f acc1 = {0.f, 0.f, 0.f, 0.f, 0.f, 0.f, 0.f, 0.f};
#pragma unroll
    for (int kk = 0; kk < 8; kk++) {
      const int K = 4 * kk + 2 * hf;          // A/B layout: VGPRv -> K = 2*half + v
      const int srow = 16 * p5 + L;           // A: lane%16 = row
      v2f a  = {Sw[srow][K], Sw[srow][K + 1]};
      v2f b0 = {M5[K][L],      M5[K + 1][L]};       // N-tile 0: cols 0..15
      v2f b1 = {M5[K][16 + L], M5[K + 1][16 + L]};  // N-tile 1: cols 16..31
      acc0 = __builtin_amdgcn_wmma_f32_16x16x4_f32(false, a, false, b0, (short)0, acc0, false, false);
      acc1 = __builtin_amdgcn_wmma_f32_16x16x4_f32(false, a, false, b1, (short)0, acc1, false, false);
    }
    // C layout: VGPR vr -> row vr+8*half, col = lane%16.  Dot with S row, reduce over 16 cols.
    float res[8];
#pragma unroll
    for (int vr = 0; vr < 8; vr++) {
      const int tr = vr + 8 * hf;
      const int srow = 16 * p5 + tr;
      float part = acc0[vr] * Sw[srow][L] + acc1[vr] * Sw[srow][16 + L];
      part += __shfl_xor(part, 1);
      part += __shfl_xor(part, 2);
      part += __shfl_xor(part, 4);
      part += __shfl_xor(part, 8);
      res[vr] = part;
    }
    if (L == 0) {
#pragma unroll
      for (int vr = 0; vr < 8; vr++)
        out[gbase + 16 * p5 + vr + 8 * hf] = MULT * res[vr];
    }
  }
}

// ============================ host launcher ================================
extern "C" void kernel_launch(void* const* d_in, const int* in_sizes, int n_in,
                              void* d_out, int out_size, void* d_ws, size_t ws_size,
                              hipStream_t stream) {
  const float* x   = (const float*)d_in[0];
  const float* th3 = (const float*)d_in[1];
  const float* th5 = (const float*)d_in[2];
  float* out = (float*)d_out;
  float* ws  = (float*)d_ws;
  const int Bv = in_sizes[0] / 13;  // 131072
  vqc_setup<<<1, 128, 0, stream>>>(th3, th5, ws);
  vqc_main<<<Bv / 256, 256, 0, stream>>>(x, ws, out);
}